// GCNLayer_35029753266585
// MI455X (gfx1250) — compile-verified
//
#include <hip/hip_runtime.h>
#include <hip/hip_bf16.h>
#include <math.h>

typedef __attribute__((ext_vector_type(2))) float v2f;
typedef __attribute__((ext_vector_type(8))) float v8f;

#define LATDIM 64
#define LEAK 0.2f

// ---------------------------------------------------------------------------
// Kernel 1: zero the aggregation buffer (d_out is poisoned by the harness).
// ---------------------------------------------------------------------------
__global__ void gcn_zero_kernel(float* __restrict__ agg, long total) {
    long i = (long)blockIdx.x * blockDim.x + threadIdx.x;
    if (i < total) agg[i] = 0.0f;
}

// ---------------------------------------------------------------------------
// Kernel 2: SpMM scatter.  64 threads per edge (2 wave32s), thread handles one
// of the 64 feature dims.  Within a wave, e is uniform (scalarized loads of
// rows/cols/vals), and the embeds gather + atomic scatter are fully coalesced
// 128B transactions.  agg (25.6 MB) and embeds (25.6 MB) are L2-resident on
// MI455X (192 MB L2), so the atomics resolve on-chip.
// ---------------------------------------------------------------------------
__global__ void gcn_spmm_kernel(const int* __restrict__ rows,
                                const int* __restrict__ cols,
                                const float* __restrict__ vals,
                                const float* __restrict__ emb,
                                float* __restrict__ agg, int n_edges) {
    long t = (long)blockIdx.x * blockDim.x + threadIdx.x;
    int e = (int)(t >> 6);
    int d = (int)(t & 63);
    if (e < n_edges) {
        int r = rows[e];
        int c = cols[e];
        float v = vals[e];
        float m = v * emb[(long)c * LATDIM + d];
        // native global_atomic_add_f32 (agg is coarse-grained device memory)
        unsafeAtomicAdd(&agg[(long)r * LATDIM + d], m);
    }
}

// ---------------------------------------------------------------------------
// Kernel 3: z = agg @ att_weight via V_WMMA_F32_16X16X4_F32.
// One wave (32 threads) per tile of 16 nodes.
//   A (16x4 f32): lanes 0-15 hold row M=lane, K = {0,1}; lanes 16-31 hold
//                 row M=lane-16, K = {2,3}   (ISA 7.12.2 layout)
//   B (4x16 f32): broadcast att_weight[k] across all 16 columns -> each
//                 lane's B values depend only on k, so layout-safe.
//   D (16x16 f32): column 0 lives in lane 0 (M=0..7 in d[0..7]) and
//                 lane 16 (M=8..15).
// 16 k-chunks of 4 cover LATDIM=64.
// ---------------------------------------------------------------------------
__global__ void gcn_matvec_wmma_kernel(const float* __restrict__ agg,
                                       const float* __restrict__ w,
                                       float* __restrict__ z, int n_nodes) {
    int tile = blockIdx.x;
    int lane = threadIdx.x;            // 0..31, EXEC all-ones for WMMA
    int half = lane >> 4;              // 0: K={0,1}, 1: K={2,3}
    int mrow = tile * 16 + (lane & 15);
    int msafe = mrow < n_nodes ? mrow : n_nodes - 1;
    int k0 = half * 2;

    const float* rowp = agg + (long)msafe * LATDIM;
    v8f d = {};
    #pragma unroll
    for (int c = 0; c < LATDIM; c += 4) {
        v2f a, b;
        a.x = rowp[c + k0];
        a.y = rowp[c + k0 + 1];
        b.x = w[c + k0];
        b.y = w[c + k0 + 1];
        d = __builtin_amdgcn_wmma_f32_16x16x4_f32(
            /*neg_a=*/false, a, /*neg_b=*/false, b,
            /*c_mod=*/(short)0, d, /*reuse_a=*/false, /*reuse_b=*/false);
    }
    // Extract column 0 of D: lane 0 -> rows tile*16+0..7, lane 16 -> +8..15
    if ((lane & 15) == 0) {
        int base = tile * 16 + half * 8;
        #pragma unroll
        for (int r = 0; r < 8; ++r) {
            int n = base + r;
            if (n < n_nodes) z[n] = d[r];
        }
    }
}

// ---------------------------------------------------------------------------
// Kernel 4/5: single-block reductions for softmax max and sum(exp(z-max)).
// 100K floats, trivially fast; single block keeps it deterministic.
// ---------------------------------------------------------------------------
__global__ void gcn_reduce_max_kernel(const float* __restrict__ z,
                                      float* __restrict__ out_max, int n) {
    __shared__ float s[1024];
    float m = -__builtin_inff();
    for (int i = threadIdx.x; i < n; i += 1024) m = fmaxf(m, z[i]);
    s[threadIdx.x] = m;
    __syncthreads();
    for (int off = 512; off > 0; off >>= 1) {
        if ((int)threadIdx.x < off)
            s[threadIdx.x] = fmaxf(s[threadIdx.x], s[threadIdx.x + off]);
        __syncthreads();
    }
    if (threadIdx.x == 0) *out_max = s[0];
}

__global__ void gcn_reduce_sumexp_kernel(const float* __restrict__ z,
                                         const float* __restrict__ mx,
                                         float* __restrict__ out_sum, int n) {
    __shared__ float s[1024];
    float m = *mx;
    float acc = 0.0f;
    for (int i = threadIdx.x; i < n; i += 1024) acc += expf(z[i] - m);
    s[threadIdx.x] = acc;
    __syncthreads();
    for (int off = 512; off > 0; off >>= 1) {
        if ((int)threadIdx.x < off)
            s[threadIdx.x] += s[threadIdx.x + off];
        __syncthreads();
    }
    if (threadIdx.x == 0) *out_sum = s[0];
}

// ---------------------------------------------------------------------------
// Kernel 6: out = leaky_relu(agg * softmax(z)), in place on d_out.
// ---------------------------------------------------------------------------
__global__ void gcn_apply_kernel(float* __restrict__ agg,
                                 const float* __restrict__ z,
                                 const float* __restrict__ mx,
                                 const float* __restrict__ sm, long total) {
    long i = (long)blockIdx.x * blockDim.x + threadIdx.x;
    if (i < total) {
        int node = (int)(i >> 6);
        float score = expf(z[node] - *mx) / *sm;
        float v = agg[i] * score;
        agg[i] = v > 0.0f ? v : LEAK * v;
    }
}

extern "C" void kernel_launch(void* const* d_in, const int* in_sizes, int n_in,
                              void* d_out, int out_size, void* d_ws, size_t ws_size,
                              hipStream_t stream) {
    const int*   rows = (const int*)d_in[0];
    const int*   cols = (const int*)d_in[1];
    const float* vals = (const float*)d_in[2];
    const float* emb  = (const float*)d_in[3];
    const float* w    = (const float*)d_in[4];

    const int n_edges = in_sizes[0];
    const int n_nodes = in_sizes[3] / LATDIM;

    float* agg = (float*)d_out;          // agg lives in d_out, finalized in place
    float* z   = (float*)d_ws;           // [n_nodes] dot products
    float* mx  = z + n_nodes;            // softmax max
    float* sm  = mx + 1;                 // softmax denom

    const long total = (long)n_nodes * LATDIM;

    gcn_zero_kernel<<<(unsigned)((total + 255) / 256), 256, 0, stream>>>(agg, total);

    long spmm_threads = (long)n_edges * LATDIM;
    gcn_spmm_kernel<<<(unsigned)((spmm_threads + 255) / 256), 256, 0, stream>>>(
        rows, cols, vals, emb, agg, n_edges);

    int n_tiles = (n_nodes + 15) / 16;
    gcn_matvec_wmma_kernel<<<n_tiles, 32, 0, stream>>>(agg, w, z, n_nodes);

    gcn_reduce_max_kernel<<<1, 1024, 0, stream>>>(z, mx, n_nodes);
    gcn_reduce_sumexp_kernel<<<1, 1024, 0, stream>>>(z, mx, sm, n_nodes);

    gcn_apply_kernel<<<(unsigned)((total + 255) / 256), 256, 0, stream>>>(
        agg, z, mx, sm, total);
}